// EncodeStateDirectly_24781961298534
// MI455X (gfx1250) — compile-verified
//
#include <hip/hip_runtime.h>
#include <hip/hip_bf16.h>

// ---------------------------------------------------------------------------
// Problem constants (from reference)
// ---------------------------------------------------------------------------
#define OBS_   64
#define HID_   84          // OBS + EXTRA(20)
#define T_     64
#define B_     16
#define L2REG  0.001f

// ws layout (floats)
#define GSZ      (64 * 64 * 84)            // 344064 : G[0..63], each 64x84 row-major
#define NTRI     2080                      // 64*65/2 lower blocks
#define LSZ      (NTRI * 4096)             // 8519680 : MMT -> L blocks, 64x64 row-major
#define WSZ      (64 * 64 * 16)            // 65536   : W / z blocks, 64x16 row-major
#define OFF_G    0
#define OFF_L    (OFF_G + GSZ)
#define OFF_W    (OFF_L + LSZ)
#define OFF_P    (OFF_W + WSZ)             // 64 partial sums

typedef __attribute__((ext_vector_type(2))) float v2f;
typedef __attribute__((ext_vector_type(8))) float v8f;

__device__ __forceinline__ size_t tri(int i, int j) {
    return (size_t)((i * (i + 1)) / 2 + j);
}

// ---------------------------------------------------------------------------
// f32 WMMA tile helper: 16x16 (MxN) accumulate of A(16xK) * B(16xK)^T
// A rows (M) at Abase with leading dim lda; B rows (N) at Bbase (i.e. the
// operand fed to the WMMA B slot is Bbase^T). K must be a multiple of 4.
// Per ISA 7.12.2: A 16x4 f32 -> lane holds row (lane&15), VGPR0/1 = K pair
// selected by lane half; C/D 16x16 f32 -> VGPR v holds M = v + 8*(lane>=16),
// lane&15 = N.
// ---------------------------------------------------------------------------
__device__ __forceinline__ v8f wmma_f32_acc(const float* Abase, int lda,
                                            const float* Bbase, int ldb,
                                            int K, int lane, v8f acc)
{
    const int half = (lane < 16) ? 0 : 2;
    const int mn   = lane & 15;
    for (int k0 = 0; k0 < K; k0 += 4) {
        v2f a, b;
        a.x = Abase[mn * lda + k0 + half + 0];
        a.y = Abase[mn * lda + k0 + half + 1];
        b.x = Bbase[mn * ldb + k0 + half + 0];
        b.y = Bbase[mn * ldb + k0 + half + 1];
        acc = __builtin_amdgcn_wmma_f32_16x16x4_f32(false, a, false, b,
                                                    (short)0, acc, false, false);
    }
    return acc;
}

// ---------------------------------------------------------------------------
// 1) G[0] = Treg = C + L2*I(64 part) ;  G[k+1] = G[k] @ A   (sequential, 1 WG)
//    A: 84x84, G[k]: 64x84 row-major. 5376 = 21*256 outputs/step.
// ---------------------------------------------------------------------------
__global__ void __launch_bounds__(256) build_g_kernel(const float* __restrict__ A,
                                                      const float* __restrict__ C,
                                                      float* __restrict__ G)
{
    __shared__ float sA[84 * 84];
    __shared__ float sG[64 * 84];
    const int tid = threadIdx.x;

    for (int e = tid; e < 84 * 84; e += 256) sA[e] = A[e];
    for (int q = 0; q < 21; ++q) {
        int e = tid + q * 256;
        int r = e / 84, c = e % 84;
        float v = C[e];
        if (c == r) v += L2REG;
        sG[e] = v;
        G[e]  = v;
    }
    __syncthreads();

    for (int step = 1; step < 64; ++step) {
        float out[21];
        for (int q = 0; q < 21; ++q) {
            int e = tid + q * 256;
            int r = e / 84, c = e % 84;
            float acc = 0.0f;
            const float* gr = &sG[r * 84];
            for (int k = 0; k < 84; ++k) acc = fmaf(gr[k], sA[k * 84 + c], acc);
            out[q] = acc;
        }
        __syncthreads();
        for (int q = 0; q < 21; ++q) {
            int e = tid + q * 256;
            sG[e] = out[q];
            G[(size_t)step * 5376 + e] = out[q];
        }
        __syncthreads();
    }
}

// ---------------------------------------------------------------------------
// 2a) P[a][b] = G[a] @ G[b]^T  (64x64, K=84) for all a>=b -> L[tri(a,b)]
//     One WG per pair; 8 waves, 2 tiles each, f32 WMMA from LDS.
// ---------------------------------------------------------------------------
__global__ void __launch_bounds__(256) p_blocks_kernel(const float* __restrict__ G,
                                                       float* __restrict__ L)
{
    const int idx = blockIdx.x;
    int a = (int)((sqrtf(8.0f * (float)idx + 1.0f) - 1.0f) * 0.5f);
    while ((a + 1) * (a + 2) / 2 <= idx) ++a;
    while (a * (a + 1) / 2 > idx)        --a;
    const int b = idx - a * (a + 1) / 2;

    __shared__ float sGa[64 * 84];
    __shared__ float sGb[64 * 84];
    const int tid = threadIdx.x;
    for (int e = tid; e < 5376; e += 256) {
        sGa[e] = G[(size_t)a * 5376 + e];
        sGb[e] = G[(size_t)b * 5376 + e];
    }
    __syncthreads();

    const int wave = tid >> 5, lane = tid & 31;
    float* out = L + tri(a, b) * 4096;
    for (int t = wave; t < 16; t += 8) {
        const int tm = t >> 2, tn = t & 3;
        v8f acc = (v8f){0.f,0.f,0.f,0.f,0.f,0.f,0.f,0.f};
        acc = wmma_f32_acc(sGa + tm * 16 * 84, 84, sGb + tn * 16 * 84, 84, 84, lane, acc);
        for (int v = 0; v < 8; ++v) {
            int m = tm * 16 + v + ((lane < 16) ? 0 : 8);
            int n = tn * 16 + (lane & 15);
            out[m * 64 + n] = acc[v];
        }
    }
}

// ---------------------------------------------------------------------------
// 2b) Prefix-sum along each block diagonal d: MMT[j+d][j] = cumsum_j P[j+d][j]
//     Register-carried per element: no memory RAW hazards.
// ---------------------------------------------------------------------------
__global__ void __launch_bounds__(256) prefix_diag_kernel(float* __restrict__ L)
{
    const int d = blockIdx.x;
    float acc[16];
    for (int q = 0; q < 16; ++q) acc[q] = 0.0f;
    for (int j = 0; j + d < 64; ++j) {
        float* blk = L + tri(j + d, j) * 4096;
        for (int q = 0; q < 16; ++q) {
            int e = threadIdx.x + q * 256;
            acc[q] += blk[e];
            blk[e]  = acc[q];
        }
    }
}

// ---------------------------------------------------------------------------
// 3a) Left-looking update for block column j:
//     L[tri(i,j)] -= sum_{k<j} L[i][k] @ L[j][k]^T     (i = j .. 63)
// ---------------------------------------------------------------------------
__global__ void __launch_bounds__(256) chol_update_kernel(float* __restrict__ L, int j)
{
    const int i = j + blockIdx.x;
    __shared__ float sA[64 * 64];
    __shared__ float sB[64 * 64];
    const int tid = threadIdx.x;
    const int wave = tid >> 5, lane = tid & 31;

    float* out = L + tri(i, j) * 4096;
    for (int t = wave; t < 16; t += 8) {
        const int tm = t >> 2, tn = t & 3;
        v8f acc = (v8f){0.f,0.f,0.f,0.f,0.f,0.f,0.f,0.f};
        for (int k = 0; k < j; ++k) {
            __syncthreads();
            const float* Aik = L + tri(i, k) * 4096;
            const float* Ljk = L + tri(j, k) * 4096;
            for (int e = tid; e < 4096; e += 256) { sA[e] = Aik[e]; sB[e] = Ljk[e]; }
            __syncthreads();
            acc = wmma_f32_acc(sA + tm * 16 * 64, 64, sB + tn * 16 * 64, 64, 64, lane, acc);
        }
        for (int v = 0; v < 8; ++v) {
            int m = tm * 16 + v + ((lane < 16) ? 0 : 8);
            int n = tn * 16 + (lane & 15);
            out[m * 64 + n] -= acc[v];
        }
    }
}

// 3b) Cholesky of the 64x64 diagonal block (1 WG, 64 threads, thread = row)
__global__ void __launch_bounds__(64) chol_diag_kernel(float* __restrict__ L, int j)
{
    __shared__ float s[64][65];
    float* blk = L + tri(j, j) * 4096;
    const int r = threadIdx.x;
    for (int c = 0; c < 64; ++c) s[r][c] = blk[r * 64 + c];
    __syncthreads();
    for (int c = 0; c < 64; ++c) {
        if (r == c) {
            float d = s[c][c];
            for (int k = 0; k < c; ++k) d = fmaf(-s[c][k], s[c][k], d);
            s[c][c] = sqrtf(d);
        }
        __syncthreads();
        if (r > c) {
            float v = s[r][c];
            for (int k = 0; k < c; ++k) v = fmaf(-s[r][k], s[c][k], v);
            s[r][c] = v / s[c][c];
        }
        __syncthreads();
    }
    for (int c = r + 1; c < 64; ++c) s[r][c] = 0.0f;
    __syncthreads();
    for (int c = 0; c < 64; ++c) blk[r * 64 + c] = s[r][c];
}

// 3c) TRSM: L[i][j] = S * L[j][j]^{-T}  (i = j+1 .. 63), row per thread
__global__ void __launch_bounds__(64) chol_trsm_kernel(float* __restrict__ L, int j)
{
    const int i = j + 1 + blockIdx.x;
    __shared__ float d[64 * 64];
    __shared__ float X[64][65];
    float* blk = L + tri(i, j) * 4096;
    const float* dj = L + tri(j, j) * 4096;
    const int r = threadIdx.x;
    for (int c = 0; c < 64; ++c) { d[r * 64 + c] = dj[r * 64 + c]; X[r][c] = blk[r * 64 + c]; }
    __syncthreads();
    for (int c = 0; c < 64; ++c) {
        float v = X[r][c];
        for (int k = 0; k < c; ++k) v = fmaf(-X[r][k], d[c * 64 + k], v);
        X[r][c] = v / d[c * 64 + c];
    }
    __syncthreads();
    for (int c = 0; c < 64; ++c) blk[r * 64 + c] = X[r][c];
}

// ---------------------------------------------------------------------------
// 4) Forward block substitution: W = L^{-1} Y^T   (Y^T[r][b] = x[b*4096+r])
// ---------------------------------------------------------------------------
__global__ void __launch_bounds__(256) forward_kernel(const float* __restrict__ x,
                                                      const float* __restrict__ L,
                                                      float* __restrict__ W)
{
    __shared__ float sL[64 * 64];
    __shared__ float sW[64 * 16];
    __shared__ float sWk[64 * 16];
    const int tid = threadIdx.x;

    for (int j = 0; j < 64; ++j) {
        for (int e = tid; e < 1024; e += 256) {
            int r = e >> 4, b = e & 15;
            sW[e] = x[(size_t)b * 4096 + j * 64 + r];
        }
        __syncthreads();
        for (int k = 0; k < j; ++k) {
            const float* Ljk = L + tri(j, k) * 4096;
            for (int e = tid; e < 4096; e += 256) sL[e] = Ljk[e];
            for (int e = tid; e < 1024; e += 256) sWk[e] = W[(size_t)k * 1024 + e];
            __syncthreads();
            for (int e = tid; e < 1024; e += 256) {
                int r = e >> 4, b = e & 15;
                float acc = 0.0f;
                for (int kk = 0; kk < 64; ++kk) acc = fmaf(sL[r * 64 + kk], sWk[kk * 16 + b], acc);
                sW[e] -= acc;
            }
            __syncthreads();
        }
        const float* Ljj = L + tri(j, j) * 4096;
        for (int e = tid; e < 4096; e += 256) sL[e] = Ljj[e];
        __syncthreads();
        if (tid < 16) {
            int b = tid;
            for (int r = 0; r < 64; ++r) {
                float v = sW[r * 16 + b];
                for (int k = 0; k < r; ++k) v = fmaf(-sL[r * 64 + k], sW[k * 16 + b], v);
                sW[r * 16 + b] = v / sL[r * 64 + r];
            }
        }
        __syncthreads();
        for (int e = tid; e < 1024; e += 256) W[(size_t)j * 1024 + e] = sW[e];
        __threadfence();
        __syncthreads();
    }
}

// ---------------------------------------------------------------------------
// 5) Backward block substitution (in place): z = L^{-T} W
// ---------------------------------------------------------------------------
__global__ void __launch_bounds__(256) backward_kernel(const float* __restrict__ L,
                                                       float* __restrict__ W)
{
    __shared__ float sL[64 * 64];
    __shared__ float sW[64 * 16];
    __shared__ float sWk[64 * 16];
    const int tid = threadIdx.x;

    for (int j = 63; j >= 0; --j) {
        for (int e = tid; e < 1024; e += 256) sW[e] = W[(size_t)j * 1024 + e];
        __syncthreads();
        for (int k = j + 1; k < 64; ++k) {
            const float* Lkj = L + tri(k, j) * 4096;
            for (int e = tid; e < 4096; e += 256) sL[e] = Lkj[e];
            for (int e = tid; e < 1024; e += 256) sWk[e] = W[(size_t)k * 1024 + e];
            __syncthreads();
            for (int e = tid; e < 1024; e += 256) {
                int r = e >> 4, b = e & 15;
                float acc = 0.0f;
                for (int kk = 0; kk < 64; ++kk) acc = fmaf(sL[kk * 64 + r], sWk[kk * 16 + b], acc);
                sW[e] -= acc;
            }
            __syncthreads();
        }
        const float* Ljj = L + tri(j, j) * 4096;
        for (int e = tid; e < 4096; e += 256) sL[e] = Ljj[e];
        __syncthreads();
        if (tid < 16) {
            int b = tid;
            for (int r = 63; r >= 0; --r) {
                float v = sW[r * 16 + b];
                for (int k = r + 1; k < 64; ++k) v = fmaf(-sL[k * 64 + r], sW[k * 16 + b], v);
                sW[r * 16 + b] = v / sL[r * 64 + r];
            }
        }
        __syncthreads();
        for (int e = tid; e < 1024; e += 256) W[(size_t)j * 1024 + e] = sW[e];
        __threadfence();
        __syncthreads();
    }
}

// ---------------------------------------------------------------------------
// 6) eps block k (k=1..63): E = sum_{i>=k} G[i-k]^T @ z[i]  (84x16), emit sum(E^2)
// ---------------------------------------------------------------------------
__global__ void __launch_bounds__(256) loss_kernel(const float* __restrict__ G,
                                                   const float* __restrict__ Z,
                                                   float* __restrict__ partial)
{
    const int k = blockIdx.x + 1;
    __shared__ float sG[64 * 84];
    __shared__ float sZ[64 * 16];
    __shared__ float red[256];
    const int tid = threadIdx.x;

    float accE[6];
    for (int q = 0; q < 6; ++q) accE[q] = 0.0f;

    for (int i = k; i < 64; ++i) {
        __syncthreads();
        const float* g = G + (size_t)(i - k) * 5376;
        for (int e = tid; e < 5376; e += 256) sG[e] = g[e];
        for (int e = tid; e < 1024; e += 256) sZ[e] = Z[(size_t)i * 1024 + e];
        __syncthreads();
        for (int q = 0; q < 6; ++q) {
            int e = tid + q * 256;
            if (e < 84 * 16) {
                int r = e >> 4, b = e & 15;
                float acc = 0.0f;
                for (int m = 0; m < 64; ++m) acc = fmaf(sG[m * 84 + r], sZ[m * 16 + b], acc);
                accE[q] += acc;
            }
        }
    }
    float local = 0.0f;
    for (int q = 0; q < 6; ++q) {
        int e = tid + q * 256;
        if (e < 84 * 16) local = fmaf(accE[q], accE[q], local);
    }
    red[tid] = local;
    __syncthreads();
    for (int s = 128; s > 0; s >>= 1) {
        if (tid < s) red[tid] += red[tid + s];
        __syncthreads();
    }
    if (tid == 0) partial[k - 1] = red[0];
}

// 7) finalize: mean over eps (16 x (63*84) elements)
__global__ void finalize_kernel(const float* __restrict__ partial, float* __restrict__ out)
{
    float s = 0.0f;
    for (int i = 0; i < 63; ++i) s += partial[i];
    out[0] = s / (16.0f * 5292.0f);
}

// ---------------------------------------------------------------------------
extern "C" void kernel_launch(void* const* d_in, const int* in_sizes, int n_in,
                              void* d_out, int out_size, void* d_ws, size_t ws_size,
                              hipStream_t stream)
{
    (void)in_sizes; (void)n_in; (void)out_size;
    const float* x = (const float*)d_in[0];   // (16,64,1,8,8) = (16,4096)
    const float* A = (const float*)d_in[1];   // 84x84
    const float* C = (const float*)d_in[2];   // 64x84
    float* ws = (float*)d_ws;
    if (ws_size < (size_t)(OFF_P + 64) * sizeof(float)) return;

    float* G = ws + OFF_G;
    float* L = ws + OFF_L;
    float* W = ws + OFF_W;
    float* P = ws + OFF_P;

    build_g_kernel<<<1, 256, 0, stream>>>(A, C, G);
    p_blocks_kernel<<<NTRI, 256, 0, stream>>>(G, L);
    prefix_diag_kernel<<<64, 256, 0, stream>>>(L);

    for (int j = 0; j < 64; ++j) {
        if (j > 0) chol_update_kernel<<<64 - j, 256, 0, stream>>>(L, j);
        chol_diag_kernel<<<1, 64, 0, stream>>>(L, j);
        if (j < 63) chol_trsm_kernel<<<63 - j, 64, 0, stream>>>(L, j);
    }

    forward_kernel<<<1, 256, 0, stream>>>(x, L, W);
    backward_kernel<<<1, 256, 0, stream>>>(L, W);
    loss_kernel<<<63, 256, 0, stream>>>(G, W, P);
    finalize_kernel<<<1, 1, 0, stream>>>(P, (float*)d_out);
}